// SimilarityNetwork_31722628448597
// MI455X (gfx1250) — compile-verified
//
#include <hip/hip_runtime.h>

typedef float v2f __attribute__((ext_vector_type(2)));
typedef float v8f __attribute__((ext_vector_type(8)));

#define NROWS   262144
#define DIMK    512
#define TEMPINV 10.0f
#define GROUPS  (NROWS / 16)            // 16384 groups of 16 rows
#define WPB     4                       // waves per block
#define BLOCKS1 2048
#define TOTWAVE (BLOCKS1 * WPB)         // 8192
#define CHUNK   128                     // k-chunk staged to LDS
#define PADSTR  132                     // 128 + 4 pad words -> conflict-free rows
#define TILEW   (16 * PADSTR)           // floats per tile buffer

// ---------------- Pass 1: cosine + exp((c-1)/T), per-wave partial sums ----------
__global__ __launch_bounds__(128)
void simnet_pass1(const float* __restrict__ z, const float* __restrict__ e,
                  float* __restrict__ out, float* __restrict__ partials) {
  __shared__ __align__(16) float lds_e[DIMK];
  __shared__ __align__(16) float tile[WPB][2][TILEW];   // double-buffered per wave

  const int tid  = threadIdx.x;
  const int lane = tid & 31;
  const int wave = tid >> 5;
  const int h    = lane >> 4;      // lane half: K split for A/B operands
  const int m16  = lane & 15;      // A-matrix row within group

  // stage e once per block
  for (int i = tid; i < DIMK; i += blockDim.x) lds_e[i] = e[i];
  __syncthreads();

  // per-wave ||e||^2 (cheap, once)
  float en2 = 0.f;
  #pragma unroll
  for (int j = 0; j < DIMK / 32; ++j) {
    float v = lds_e[lane + 32 * j];
    en2 = fmaf(v, v, en2);
  }
  #pragma unroll
  for (int off = 16; off > 0; off >>= 1) en2 += __shfl_xor(en2, off, 32);
  const float inv_en = rsqrtf(en2);

  float* myTile = &tile[wave][0][0];
  // 32-bit LDS byte offset of this wave's tile (flat LDS addr: low 32 = offset)
  const unsigned ldsBase = (unsigned)(uintptr_t)myTile + (unsigned)lane * 16u;
  const int gwave = blockIdx.x * WPB + wave;
  float usum = 0.f;

  for (int g = gwave; g < GROUPS; g += TOTWAVE) {
    const int r0 = g << 4;
    const int gs = __builtin_amdgcn_readfirstlane(g);        // wave-uniform
    const float* rowBase = z + (size_t)(gs << 4) * DIMK;     // saddr (SGPR pair)

    // issue one 16x128 chunk as 16 async b128 copies (HBM -> LDS, no VGPR data)
    auto issue_chunk = [&](int buf, int c0) {
      #pragma unroll
      for (int r = 0; r < 16; ++r) {
        unsigned voff = (unsigned)(r * (DIMK * 4) + c0 * 4) + (unsigned)lane * 16u;
        unsigned ldst = ldsBase + (unsigned)(buf * (TILEW * 4) + r * (PADSTR * 4));
        asm volatile("global_load_async_to_lds_b128 %0, %1, %2"
                     :: "v"(ldst), "v"(voff), "s"(rowBase) : "memory");
      }
    };

    issue_chunk(0, 0);                 // prefetch first chunk
    v8f acc = {};                      // 16x16 f32 accumulator (all columns equal)
    float sq0 = 0.f, sq1 = 0.f;        // split ||row||^2 FMA chains

    #pragma unroll
    for (int c = 0; c < DIMK / CHUNK; ++c) {
      const int c0 = c * CHUNK;
      if (c < DIMK / CHUNK - 1) {
        issue_chunk((c + 1) & 1, c0 + CHUNK);                // prefetch next
        asm volatile("s_wait_asynccnt 0x10" ::: "memory");   // current buf ready
      } else {
        asm volatile("s_wait_asynccnt 0x0" ::: "memory");
      }
      const float* tbuf = myTile + (c & 1) * TILEW;
      // 32 WMMA k-steps (K=4 each) over this chunk
      #pragma unroll 8
      for (int s = 0; s < CHUNK / 4; ++s) {
        const int kc = 4 * s + 2 * h;
        v2f a = *(const v2f*)(tbuf + m16 * PADSTR + kc);     // A: z rows
        v2f b = *(const v2f*)(lds_e + c0 + kc);              // B: e bcast all N
        sq0 = fmaf(a.x, a.x, sq0);
        sq1 = fmaf(a.y, a.y, sq1);
        acc = __builtin_amdgcn_wmma_f32_16x16x4_f32(
            false, a, false, b, (short)0, acc, false, false);
      }
    }

    // lanes l / l+16 hold complementary-k squares of row l%16
    float sq = sq0 + sq1;
    float sqrow = sq + __shfl_xor(sq, 16, 32);   // lanes 0..15: ||z_{r0+l}||^2

    // D vgpr m, lanes 0-15 -> row m ; lanes 16-31 -> row m+8 (cols identical)
    const int msel = lane & 7;
    float x0 = (msel & 1) ? acc[1] : acc[0];
    float x1 = (msel & 1) ? acc[3] : acc[2];
    float x2 = (msel & 1) ? acc[5] : acc[4];
    float x3 = (msel & 1) ? acc[7] : acc[6];
    float y0 = (msel & 2) ? x1 : x0;
    float y1 = (msel & 2) ? x3 : x2;
    float dsel = (msel & 4) ? y1 : y0;           // lane l: dot(row (l&7)+8h)
    // route so lanes 0..15 carry dot(row lane)
    int src = ((lane & 8) ? (lane + 8) : lane) & 31;
    float dot = __shfl(dsel, src, 32);

    float c = dot * rsqrtf(sqrow) * inv_en;      // cosine in [-1,1]
    float u = __expf((c - 1.0f) * TEMPINV);      // analytic max shift (cos<=1)
    if (lane < 16) out[r0 + lane] = u;
    usum += (lane < 16) ? u : 0.f;
  }

  #pragma unroll
  for (int off = 16; off > 0; off >>= 1) usum += __shfl_xor(usum, off, 32);
  if (lane == 0) partials[gwave] = usum;
}

// ---------------- Pass 2: deterministic reduction -> 1/sum ----------------------
__global__ __launch_bounds__(256)
void simnet_reduce(float* __restrict__ partials) {
  __shared__ float red[256];
  float s = 0.f;
  for (int p = threadIdx.x; p < TOTWAVE; p += 256) s += partials[p];
  red[threadIdx.x] = s;
  __syncthreads();
  for (int off = 128; off > 0; off >>= 1) {
    if (threadIdx.x < off) red[threadIdx.x] += red[threadIdx.x + off];
    __syncthreads();
  }
  if (threadIdx.x == 0) partials[TOTWAVE] = 1.0f / red[0];
}

// ---------------- Pass 3: scale in place (kernel1 rewrites out each call) -------
__global__ __launch_bounds__(256)
void simnet_scale(float* __restrict__ out, const float* __restrict__ partials) {
  const float inv = partials[TOTWAVE];
  const int i = blockIdx.x * 256 + threadIdx.x;   // N % 256 == 0
  out[i] *= inv;
}

extern "C" void kernel_launch(void* const* d_in, const int* in_sizes, int n_in,
                              void* d_out, int out_size, void* d_ws, size_t ws_size,
                              hipStream_t stream) {
  (void)in_sizes; (void)n_in; (void)out_size; (void)ws_size;
  const float* z = (const float*)d_in[0];
  const float* e = (const float*)d_in[1];
  float* out = (float*)d_out;
  float* ws  = (float*)d_ws;   // [0..TOTWAVE-1] partial sums, [TOTWAVE] = 1/sum

  simnet_pass1<<<BLOCKS1, 128, 0, stream>>>(z, e, out, ws);
  simnet_reduce<<<1, 256, 0, stream>>>(ws);
  simnet_scale<<<NROWS / 256, 256, 0, stream>>>(out, ws);
}